// NearestHardSpikeLoss_17059610099997
// MI455X (gfx1250) — compile-verified
//
#include <hip/hip_runtime.h>

#define B_ 16
#define F_ 256
#define T_ 4096
#define H_ 8
#define BT (B_ * T_)

typedef __attribute__((ext_vector_type(2))) float v2f;
typedef __attribute__((ext_vector_type(8))) float v8f;

// ---------------------------------------------------------------------------
// Kernel 0: zero the per-neuron spike-in-window counters (atomic accumulators)
// ---------------------------------------------------------------------------
__global__ void init_ws(unsigned* __restrict__ a_num) {
  a_num[threadIdx.x] = 0u;
}

// ---------------------------------------------------------------------------
// Kernel 1: single streaming pass over `spikes` (the only mandatory HBM read).
// Produces per-(b,t) "any spike in f-chunk" flags and a_num[f] = sum spikes*N.
// grid = (T/1024, B, F/64), block = 256 threads, each thread owns 4 t's.
// ---------------------------------------------------------------------------
__global__ void spike_pass(const float* __restrict__ spikes,
                           const float* __restrict__ y,
                           unsigned* __restrict__ a_num,
                           unsigned char* __restrict__ any8) {
  const int tid = threadIdx.x;
  const int t0  = blockIdx.x * 1024;
  const int b   = blockIdx.y;
  const int fc  = blockIdx.z;

  __shared__ float    ysh[1024 + 2 * H_];
  __shared__ unsigned af[64];
  if (tid < 64) af[tid] = 0u;

  // y halo tile: ysh[i] corresponds to global time t0 - H + i (0 outside range)
  for (int i = tid; i < 1024 + 2 * H_; i += 256) {
    int t = t0 - H_ + i;
    ysh[i] = (t >= 0 && t < T_) ? y[b * T_ + t] : 0.0f;
  }
  __syncthreads();

  // N (dilated event mask) for this thread's 4 time steps
  const int tl = tid * 4;
  int n0 = 0, n1 = 0, n2 = 0, n3 = 0;
  for (int w = 0; w < 2 * H_ + 1; ++w) {
    n0 |= ysh[tl + 0 + w] > 0.5f;
    n1 |= ysh[tl + 1 + w] > 0.5f;
    n2 |= ysh[tl + 2 + w] > 0.5f;
    n3 |= ysh[tl + 3 + w] > 0.5f;
  }

  const float4* sp = reinterpret_cast<const float4*>(
      spikes + (size_t)(b * F_ + fc * 64) * T_ + t0);
  const int rs = T_ / 4;
  unsigned c0 = 0, c1 = 0, c2 = 0, c3 = 0;
  for (int fl = 0; fl < 64; ++fl) {
    float4 s = sp[(size_t)fl * rs + tid];       // 512B per wave32, coalesced
    int s0 = s.x > 0.5f, s1 = s.y > 0.5f, s2 = s.z > 0.5f, s3 = s.w > 0.5f;
    c0 += (unsigned)s0; c1 += (unsigned)s1; c2 += (unsigned)s2; c3 += (unsigned)s3;
    unsigned hits = (unsigned)(s0 & n0) + (unsigned)(s1 & n1) +
                    (unsigned)(s2 & n2) + (unsigned)(s3 & n3);
    if (hits) atomicAdd(&af[fl], hits);         // integer -> order independent
  }

  // per-(t, f-chunk) any-spike flags, packed 4 bytes per thread (no atomics)
  unsigned pack = (c0 ? 1u : 0u) | ((c1 ? 1u : 0u) << 8) |
                  ((c2 ? 1u : 0u) << 16) | ((c3 ? 1u : 0u) << 24);
  *reinterpret_cast<unsigned*>(any8 + (size_t)fc * BT + (size_t)b * T_ + t0 + tl) = pack;

  __syncthreads();
  if (tid < 64) {
    unsigned v = af[tid];
    if (v) atomicAdd(&a_num[fc * 64 + tid], v); // integer -> deterministic
  }
}

// ---------------------------------------------------------------------------
// Kernel 2: per-(b,t) window logic. v_hist is only touched by sparse gathers.
// grid = BT/256 blocks of 256 threads; per-block fixed-order tree reduction.
// ---------------------------------------------------------------------------
__global__ void window_pass(const float* __restrict__ spikes,
                            const float* __restrict__ v_hist,
                            const float* __restrict__ theta,
                            const float* __restrict__ y,
                            const unsigned char* __restrict__ any8,
                            float* __restrict__ partials) {
  const int tid = threadIdx.x;
  const int idx = blockIdx.x * 256 + tid;
  const int b = idx >> 12;          // T_ == 4096
  const int t = idx & (T_ - 1);

  const bool has_event = y[idx] > 0.5f;
  const int lo = (t - H_ < 0) ? 0 : t - H_;
  const int hi = (t + H_ > T_ - 1) ? T_ - 1 : t + H_;

  int N = 0;
  for (int tau = lo; tau <= hi; ++tau) N |= y[b * T_ + tau] > 0.5f;

  auto asp = [&](int tau) -> bool {
    size_t o = (size_t)b * T_ + tau;
    return (any8[o] | any8[o + (size_t)BT] | any8[o + 2 * (size_t)BT] |
            any8[o + 3 * (size_t)BT]) != 0;
  };

  // nearest spike time; ties at equal distance resolve to the earlier time,
  // matching argmin over dt = -H..H of the masked |dt| array.
  int t_star = -1;
  if (asp(t)) t_star = t;
  else {
    for (int d = 1; d <= H_; ++d) {
      if (t - d >= 0 && asp(t - d))      { t_star = t - d; break; }
      if (t + d <= T_ - 1 && asp(t + d)) { t_star = t + d; break; }
    }
  }

  float push_s = 0.f, push_n = 0.f, miss_s = 0.f, miss_n = 0.f;

  if (has_event && t_star >= 0) {            // push term (sparse gather)
    int fs = 0;
    const float* sp = spikes + (size_t)b * F_ * T_ + t_star;
    for (int f = 0; f < F_; ++f) {
      if (sp[(size_t)f * T_] > 0.5f) { fs = f; break; }
    }
    float z = v_hist[((size_t)b * F_ + fs) * T_ + t_star] - theta[fs];
    float r = 0.2f - z;
    push_s = r > 0.f ? r : 0.f;
    push_n = 1.f;
  }

  if (has_event && t_star < 0) {             // miss term (extremely rare)
    float zm = -3.4e38f;
    const float* vb = v_hist + (size_t)b * F_ * T_;
    for (int f = 0; f < F_; ++f) {
      float th = theta[f];
      const float* row = vb + (size_t)f * T_;
      for (int tau = lo; tau <= hi; ++tau) {
        float z = row[tau] - th;
        zm = zm > z ? zm : z;
      }
    }
    float r = 0.2f - zm;
    miss_s = r > 0.f ? r : 0.f;
    miss_n = 1.f;
  }

  float outside = N ? 0.f : 1.f;
  float fp_num  = asp(t) ? outside : 0.f;
  float fp_den  = outside;                   // sum_N = B*T - sum(fp_den)

  __shared__ float sred[256];
  float vals[6] = {push_s, push_n, miss_s, miss_n, fp_num, fp_den};
  for (int j = 0; j < 6; ++j) {
    __syncthreads();
    sred[tid] = vals[j];
    __syncthreads();
    for (int s = 128; s > 0; s >>= 1) {
      if (tid < s) sred[tid] += sred[tid + s];
      __syncthreads();
    }
    if (tid == 0) partials[blockIdx.x * 8 + j] = sred[0];
  }
}

// ---------------------------------------------------------------------------
// Kernel 3: one wave (EXEC all ones, as WMMA requires). Deterministic serial
// sum of block partials, then mean/variance of a_f[256] on the matrix pipe
// via V_WMMA_F32_16X16X4_F32 with an all-ones B (exact f32 row sums).
// ---------------------------------------------------------------------------
__global__ void finalize(const unsigned* __restrict__ a_num,
                         const float* __restrict__ partials,
                         float* __restrict__ out) {
  const int lane = threadIdx.x;              // 0..31
  __shared__ float sums[6];
  if (lane == 0) {
    float s0 = 0, s1 = 0, s2 = 0, s3 = 0, s4 = 0, s5 = 0;
    for (int i = 0; i < 256; ++i) {
      const float* p = partials + i * 8;
      s0 += p[0]; s1 += p[1]; s2 += p[2]; s3 += p[3]; s4 += p[4]; s5 += p[5];
    }
    sums[0] = s0; sums[1] = s1; sums[2] = s2;
    sums[3] = s3; sums[4] = s4; sums[5] = s5;
  }
  __syncthreads();
  const float push_s = sums[0], push_n = sums[1];
  const float miss_s = sums[2], miss_n = sums[3];
  const float fp_num = sums[4], fp_den = sums[5];
  const float sum_N  = (float)(B_ * T_) - fp_den;
  const float denom  = sum_N + 1e-6f;

  // A-matrix 16x4 f32 layout: lane m (0..15) holds K=0,1; lanes 16..31 K=2,3.
  const int m  = lane & 15;
  const int kb = (lane >> 4) * 2;
  float av[4][2];
  for (int c = 0; c < 4; ++c) {
    int i0 = 64 * c + 4 * m + kb;
    av[c][0] = (float)a_num[i0]     / denom;
    av[c][1] = (float)a_num[i0 + 1] / denom;
  }

  v2f ones = {1.0f, 1.0f};
  v8f acc1 = {0, 0, 0, 0, 0, 0, 0, 0};
  for (int c = 0; c < 4; ++c) {
    v2f a = {av[c][0], av[c][1]};
    acc1 = __builtin_amdgcn_wmma_f32_16x16x4_f32(false, a, false, ones,
                                                 (short)0, acc1, false, false);
  }
  // D[m,n] = row-sum of A (identical across n); lanes 0/16 hold rows 0-7/8-15
  float loc1 = acc1[0] + acc1[1] + acc1[2] + acc1[3] +
               acc1[4] + acc1[5] + acc1[6] + acc1[7];
  float S1   = __shfl(loc1, 0, 32) + __shfl(loc1, 16, 32);
  float mean = S1 * (1.0f / 256.0f);

  v8f acc2 = {0, 0, 0, 0, 0, 0, 0, 0};
  for (int c = 0; c < 4; ++c) {
    float d0 = av[c][0] - mean, d1 = av[c][1] - mean;
    v2f a = {d0 * d0, d1 * d1};
    acc2 = __builtin_amdgcn_wmma_f32_16x16x4_f32(false, a, false, ones,
                                                 (short)0, acc2, false, false);
  }
  float loc2 = acc2[0] + acc2[1] + acc2[2] + acc2[3] +
               acc2[4] + acc2[5] + acc2[6] + acc2[7];
  float S2 = __shfl(loc2, 0, 32) + __shfl(loc2, 16, 32);
  const float L_homeo = S2 * (1.0f / 256.0f);

  const float L_event = push_n > 0.f ? push_s / fmaxf(push_n, 1.f) : 0.f;
  const float L_miss  = miss_n > 0.f ? miss_s / fmaxf(miss_n, 1.f) : 0.f;
  const float L_fp    = fp_num / (fp_den + 1e-6f);

  if (lane == 0)
    out[0] = L_event + 1.0f * L_miss + 0.2f * L_fp + 0.05f * L_homeo;
}

// ---------------------------------------------------------------------------
extern "C" void kernel_launch(void* const* d_in, const int* in_sizes, int n_in,
                              void* d_out, int out_size, void* d_ws, size_t ws_size,
                              hipStream_t stream) {
  (void)in_sizes; (void)n_in; (void)out_size; (void)ws_size;
  const float* spikes = (const float*)d_in[0];
  const float* v_hist = (const float*)d_in[1];
  const float* theta  = (const float*)d_in[2];
  const float* y      = (const float*)d_in[3];
  float* out = (float*)d_out;

  // workspace layout: [0,1024) a_num u32[256]; [1024,9216) partials f32[256*8];
  // [16384, 16384+4*BT) any-spike flags (4 f-chunks x B*T bytes)
  unsigned*      a_num    = (unsigned*)d_ws;
  float*         partials = (float*)((char*)d_ws + 1024);
  unsigned char* any8     = (unsigned char*)d_ws + 16384;

  hipLaunchKernelGGL(init_ws, dim3(1), dim3(256), 0, stream, a_num);
  hipLaunchKernelGGL(spike_pass, dim3(T_ / 1024, B_, F_ / 64), dim3(256), 0,
                     stream, spikes, y, a_num, any8);
  hipLaunchKernelGGL(window_pass, dim3(BT / 256), dim3(256), 0, stream,
                     spikes, v_hist, theta, y, any8, partials);
  hipLaunchKernelGGL(finalize, dim3(1), dim3(32), 0, stream,
                     a_num, partials, out);
}